// GravNet_29420525978173
// MI455X (gfx1250) — compile-verified
//
#include <hip/hip_runtime.h>
#include <math.h>

constexpr int NB    = 16;     // batches
constexpr int NV    = 2048;   // vertices
constexpr int NF    = 64;     // input features
constexpr int NDIM  = 4;      // coord dims
constexpr int NPROP = 22;     // propagated feats
constexpr int KNEI  = 40;     // top-k incl. self
constexpr int KOUT  = 39;     // neighbors kept
constexpr int NFILT = 64;     // output filters
constexpr int KCAT  = NF + 2 * NPROP;   // 108, divisible by 4

typedef float v2f __attribute__((ext_vector_type(2)));
typedef float v8f __attribute__((ext_vector_type(8)));

// ---------------------------------------------------------------------------
// Kernel 1: coords = x @ W_s + b_s ; feats = x @ W_flr + b_flr
// One thread per row; weights staged in LDS. Tiny GEMMs (~110 MFLOP total).
// ---------------------------------------------------------------------------
__global__ __launch_bounds__(256) void k_embed(
    const float* __restrict__ x,
    const float* __restrict__ Ws, const float* __restrict__ bs,
    const float* __restrict__ Wf, const float* __restrict__ bf,
    float* __restrict__ coords, float* __restrict__ feats)
{
    __shared__ float sWs[NF * NDIM];
    __shared__ float sWf[NF * NPROP];
    int tid = threadIdx.x;
    for (int i = tid; i < NF * NDIM;  i += 256) sWs[i] = Ws[i];
    for (int i = tid; i < NF * NPROP; i += 256) sWf[i] = Wf[i];
    __syncthreads();

    int row = blockIdx.x * 256 + tid;            // 0 .. NB*NV-1
    const float* xr = x + (size_t)row * NF;
    float xl[NF];
    #pragma unroll
    for (int k = 0; k < NF; ++k) xl[k] = xr[k];

    #pragma unroll
    for (int j = 0; j < NDIM; ++j) {
        float acc = bs[j];
        for (int k = 0; k < NF; ++k) acc = fmaf(xl[k], sWs[k * NDIM + j], acc);
        coords[(size_t)row * NDIM + j] = acc;
    }
    for (int j = 0; j < NPROP; ++j) {
        float acc = bf[j];
        for (int k = 0; k < NF; ++k) acc = fmaf(xl[k], sWf[k * NPROP + j], acc);
        feats[(size_t)row * NPROP + j] = acc;
    }
}

// ---------------------------------------------------------------------------
// Kernel 2: per-query top-40 nearest neighbors (squared distance in 4-D).
// Whole batch's candidate coords staged in LDS (32 KB). Top-K kept in
// registers via fully-unrolled predicated insertion sort (no dynamic index,
// no scratch). Entry 0 (self, d=0) dropped on output, matching the reference.
// ---------------------------------------------------------------------------
__global__ __launch_bounds__(256) void k_knn(
    const float* __restrict__ coords,
    int* __restrict__ nidx, float* __restrict__ wgt)
{
    __shared__ float4 sc[NV];                    // 32 KB
    int tid = threadIdx.x;
    int b     = blockIdx.x >> 3;                 // 8 blocks per batch
    int qbase = (blockIdx.x & 7) * 256;
    const float4* cb = (const float4*)(coords + (size_t)b * NV * NDIM);
    for (int i = tid; i < NV; i += 256) sc[i] = cb[i];
    __syncthreads();

    int q = qbase + tid;
    float4 me = sc[q];

    float kd[KNEI];
    int   ki[KNEI];
    #pragma unroll
    for (int k = 0; k < KNEI; ++k) { kd[k] = 3.0e38f; ki[k] = 0; }

    for (int c = 0; c < NV; ++c) {
        float4 o = sc[c];
        float dx = me.x - o.x, dy = me.y - o.y;
        float dz = me.z - o.z, dw = me.w - o.w;
        float d = dx * dx + dy * dy + dz * dz + dw * dw;
        if (d < kd[KNEI - 1]) {
            bool placed = false;
            #pragma unroll
            for (int k = KNEI - 1; k >= 1; --k) {
                bool sh = !placed && (kd[k - 1] > d);
                if (sh)           { kd[k] = kd[k - 1]; ki[k] = ki[k - 1]; }
                else if (!placed) { kd[k] = d; ki[k] = c; placed = true; }
            }
            if (!placed) { kd[0] = d; ki[0] = c; }
        }
    }

    size_t row = (size_t)b * NV + q;
    #pragma unroll
    for (int k = 1; k < KNEI; ++k) {
        nidx[row * KOUT + (k - 1)] = ki[k];
        wgt [row * KOUT + (k - 1)] = expf(-fabsf(10.0f * kd[k]));
    }
}

// ---------------------------------------------------------------------------
// Kernel 3a: gather neighbor feats, weight, max+mean pool, concat with x.
// updated[row] = [ x(64) | max(22) | mean(22) ]   (KCAT = 108)
// ---------------------------------------------------------------------------
__global__ __launch_bounds__(256) void k_pool(
    const float* __restrict__ x,
    const float* __restrict__ feats,
    const int*   __restrict__ nidx,
    const float* __restrict__ wgt,
    float* __restrict__ upd)
{
    int row = blockIdx.x * 256 + threadIdx.x;
    int b = row / NV;
    const float* fb = feats + (size_t)b * NV * NPROP;

    float mx[NPROP], sm[NPROP];
    #pragma unroll
    for (int f = 0; f < NPROP; ++f) { mx[f] = -3.0e38f; sm[f] = 0.0f; }

    for (int n = 0; n < KOUT; ++n) {
        int   j = nidx[(size_t)row * KOUT + n];
        float w = wgt [(size_t)row * KOUT + n];
        const float* fr = fb + (size_t)j * NPROP;
        #pragma unroll
        for (int f = 0; f < NPROP; ++f) {
            float v = fr[f] * w;
            mx[f] = fmaxf(mx[f], v);
            sm[f] += v;
        }
    }

    float* ur = upd + (size_t)row * KCAT;
    const float* xr = x + (size_t)row * NF;
    for (int k = 0; k < NF; ++k) ur[k] = xr[k];
    #pragma unroll
    for (int f = 0; f < NPROP; ++f) {
        ur[NF + f]         = mx[f];
        ur[NF + NPROP + f] = sm[f] * (1.0f / (float)KOUT);
    }
}

// ---------------------------------------------------------------------------
// Kernel 3b: out = tanh(updated @ W_out + b_out) via V_WMMA_F32_16X16X4_F32.
// One wave per 16-row M-tile, looping 4 N-tiles; 27 K-steps of 4.
// A frag (16x4 f32): lane = M (mod 16); v0/v1 = K0/K1 (lanes 0-15),
//                    K2/K3 (lanes 16-31).  B frag mirrored (lane = N).
// C/D frag: lane = N (mod 16); VGPR r = row r (lanes 0-15) / row 8+r.
// EXEC is all-ones everywhere around the WMMA (uniform control flow).
// ---------------------------------------------------------------------------
__global__ __launch_bounds__(128) void k_out(
    const float* __restrict__ upd,
    const float* __restrict__ Wo, const float* __restrict__ bo,
    float* __restrict__ out)
{
    __shared__ float sW[KCAT * NFILT];           // 27.6 KB
    __shared__ float sB[NFILT];
    int tid = threadIdx.x;
    for (int i = tid; i < KCAT * NFILT; i += 128) sW[i] = Wo[i];
    if (tid < NFILT) sB[tid] = bo[tid];
    __syncthreads();

    int wave = tid >> 5;
    int lane = tid & 31;
    int l16  = lane & 15;
    int hi   = lane >> 4;                        // 0 = lanes 0-15, 1 = 16-31
    int mtile = blockIdx.x * 4 + wave;           // 2048 M-tiles total
    const float* arp = upd + (size_t)(mtile * 16 + l16) * KCAT;

    for (int nt = 0; nt < 4; ++nt) {
        float bias = sB[nt * 16 + l16];
        v8f c;
        #pragma unroll
        for (int i = 0; i < 8; ++i) c[i] = bias;

        #pragma unroll
        for (int ks = 0; ks < KCAT / 4; ++ks) {  // 27 WMMA steps
            int k0 = ks * 4 + hi * 2;
            v2f a;
            a[0] = arp[k0];
            a[1] = arp[k0 + 1];
            v2f bmat;
            bmat[0] = sW[(k0    ) * NFILT + nt * 16 + l16];
            bmat[1] = sW[(k0 + 1) * NFILT + nt * 16 + l16];
            c = __builtin_amdgcn_wmma_f32_16x16x4_f32(
                    /*neg_a=*/false, a, /*neg_b=*/false, bmat,
                    /*c_mod=*/(short)0, c, /*reuse_a=*/false, /*reuse_b=*/false);
        }

        int col   = nt * 16 + l16;
        int rbase = mtile * 16 + hi * 8;
        #pragma unroll
        for (int r = 0; r < 8; ++r)
            out[(size_t)(rbase + r) * NFILT + col] = tanhf(c[r]);
    }
}

// ---------------------------------------------------------------------------
extern "C" void kernel_launch(void* const* d_in, const int* in_sizes, int n_in,
                              void* d_out, int out_size, void* d_ws, size_t ws_size,
                              hipStream_t stream)
{
    const float* x  = (const float*)d_in[0];
    const float* Ws = (const float*)d_in[1];
    const float* bs = (const float*)d_in[2];
    const float* Wf = (const float*)d_in[3];
    const float* bf = (const float*)d_in[4];
    const float* Wo = (const float*)d_in[5];
    const float* bo = (const float*)d_in[6];
    float* out = (float*)d_out;

    // workspace carve-out (~28 MB, all 4-byte aligned)
    float* coords = (float*)d_ws;                              // NB*NV*4
    float* feats  = coords + (size_t)NB * NV * NDIM;           // NB*NV*22
    int*   nidx   = (int*)(feats + (size_t)NB * NV * NPROP);   // NB*NV*39
    float* wgt    = (float*)(nidx + (size_t)NB * NV * KOUT);   // NB*NV*39
    float* upd    = wgt + (size_t)NB * NV * KOUT;              // NB*NV*108

    k_embed<<<dim3((NB * NV) / 256), dim3(256), 0, stream>>>(x, Ws, bs, Wf, bf,
                                                             coords, feats);
    k_knn  <<<dim3(NB * 8),          dim3(256), 0, stream>>>(coords, nidx, wgt);
    k_pool <<<dim3((NB * NV) / 256), dim3(256), 0, stream>>>(x, feats, nidx, wgt, upd);
    k_out  <<<dim3((NB * NV) / 64),  dim3(128), 0, stream>>>(upd, Wo, bo, out);

    (void)in_sizes; (void)n_in; (void)out_size; (void)ws_size;
}